// Ani_19370302505420
// MI455X (gfx1250) — compile-verified
//
#include <hip/hip_runtime.h>
#include <math.h>

// ---------------------------------------------------------------------------
// ANI-1x AEV (single species) for N=256 atoms on gfx1250 (MI455X).
//   Pass 1: distance matrix via V_WMMA_F32_16X16X4_F32 Gram-matrix tiles.
//   Pass 2: per-atom fused radial + angular AEV with LDS neighbor lists.
// ---------------------------------------------------------------------------

typedef __attribute__((ext_vector_type(2))) float v2f;
typedef __attribute__((ext_vector_type(8))) float v8f;

#define N_ATOMS 256
#define RCR 5.2f
#define RCA 3.5f
#define PI_F 3.14159265358979f
#define MAXNB 64

// ---------------------------------------------------------------------------
// Kernel 1: dist[i,j] = sqrt(|xi|^2 + |xj|^2 - 2 xi.xj + (i==j))
// One wave per 16x16 tile. A = rows (16x4, K=xyz0), B = cols (4x16).
// f32 A-matrix 16x4 layout (ISA 7.12.2): lanes 0-15 hold K=0,1 in v0,v1 for
// M=lane; lanes 16-31 hold K=2,3 for M=lane-16. B mirrors with N=lane%16.
// ---------------------------------------------------------------------------
__global__ __launch_bounds__(32) void ani_dist_wmma(const float* __restrict__ xyz,
                                                    float* __restrict__ dist) {
    const int lane = threadIdx.x;
    const int half = lane >> 4;      // 0: K=0,1   1: K=2,3
    const int lm   = lane & 15;
    const int rowBase = blockIdx.y * 16;
    const int colBase = blockIdx.x * 16;

    const float rx = xyz[(rowBase + lm) * 3 + 0];
    const float ry = xyz[(rowBase + lm) * 3 + 1];
    const float rz = xyz[(rowBase + lm) * 3 + 2];
    const float cx = xyz[(colBase + lm) * 3 + 0];
    const float cy = xyz[(colBase + lm) * 3 + 1];
    const float cz = xyz[(colBase + lm) * 3 + 2];
    const float nrow = rx * rx + ry * ry + rz * rz;   // |x_row(lm)|^2, in lane lm and lm+16
    const float ncol = cx * cx + cy * cy + cz * cz;   // |x_col(lm)|^2

    v2f a, b;
    a.x = half ? rz : rx;   a.y = half ? 0.0f : ry;   // K=3 padded with 0
    b.x = half ? cz : cx;   b.y = half ? 0.0f : cy;
    v8f c = {};
    // D = A(16x4) x B(4x16) + 0 : Gram tile of coordinates
    c = __builtin_amdgcn_wmma_f32_16x16x4_f32(false, a, false, b, (short)0, c, false, false);

    // C/D layout: VGPR r, lane l -> element (M = r + 8*(l>>4), N = l&15)
#pragma unroll
    for (int r = 0; r < 8; ++r) {
        const int   M  = r + 8 * half;
        const float nI = __shfl(nrow, M, 32);         // row-norm lives in lane M
        const int   gi = rowBase + M;
        const int   gj = colBase + lm;
        float d2 = nI + ncol - 2.0f * c[r];
        d2 += (gi == gj) ? 1.0f : 0.0f;               // reference pads diagonal with eye
        dist[gi * N_ATOMS + gj] = sqrtf(fmaxf(d2, 0.0f));
    }
}

// ---------------------------------------------------------------------------
// Kernel 2: one 256-thread block per atom i. Thread t handles atom j=t.
//   radial:   16 LDS accumulators (ds_add_f32), ~19/256 threads in cutoff
//   angular:  compact neighbors (<3.5A) to LDS, stride threads over ordered
//             pairs (j,k), j!=k; 4x8 LDS accumulators.
// Avoids acos/pow: ct=0.95*cos, st=sqrt(1-ct^2), cos(t-s)=ct*cs+st*ss,
// v^32 = 5 squarings.
// ---------------------------------------------------------------------------
__global__ __launch_bounds__(256) void ani_aev(const float* __restrict__ xyz,
                                               const float* __restrict__ q,
                                               const float* __restrict__ dist,
                                               float* __restrict__ out) {
    __shared__ float s_rad[16];
    __shared__ float s_ang[32];
    __shared__ int   s_cnt;
    __shared__ float nb_d[MAXNB], nb_w[MAXNB];
    __shared__ float nb_x[MAXNB], nb_y[MAXNB], nb_z[MAXNB];

    const int i = blockIdx.x;
    const int t = threadIdx.x;
    if (t < 16) s_rad[t] = 0.0f;
    if (t < 32) s_ang[t] = 0.0f;
    if (t == 0) s_cnt = 0;
    __syncthreads();

    const float xi = xyz[i * 3 + 0];
    const float yi = xyz[i * 3 + 1];
    const float zi = xyz[i * 3 + 2];

    const int   j  = t;
    const float d  = dist[i * N_ATOMS + j];
    const float qj = q[j];

    // ---- radial term of atom j ----
    if (j != i && d < RCR) {
        const float fc   = 0.5f * __cosf(PI_F * d * (1.0f / RCR)) + 0.5f;
        const float base = 0.25f * fc * qj;
#pragma unroll
        for (int m = 0; m < 16; ++m) {
            const float s  = 0.9f + 0.26875f * (float)m;
            const float dm = d - s;
            atomicAdd(&s_rad[m], base * __expf(-16.0f * dm * dm));
        }
    }

    // ---- angular neighbor compaction ----
    if (j != i && d < RCA) {
        const float fca = 0.5f * __cosf(PI_F * d * (1.0f / RCA)) + 0.5f;
        const int idx = atomicAdd(&s_cnt, 1);
        if (idx < MAXNB) {
            nb_d[idx] = d;
            nb_w[idx] = fca * qj;
            nb_x[idx] = xyz[j * 3 + 0] - xi;
            nb_y[idx] = xyz[j * 3 + 1] - yi;
            nb_z[idx] = xyz[j * 3 + 2] - zi;
        }
    }
    __syncthreads();
    const int cnt = min(s_cnt, MAXNB);

    // angle-shift cos/sin, fully unrolled -> stays in registers
    float czv[8], szv[8];
#pragma unroll
    for (int z = 0; z < 8; ++z) {
        const float shf = PI_F / 16.0f + (PI_F / 8.0f) * (float)z;
        __sincosf(shf, &szv[z], &czv[z]);
    }

    // ---- ordered-pair loop (reference double-counts ordered pairs, net x1) ----
    const int npairs = cnt * cnt;
    for (int p = t; p < npairs; p += 256) {
        const int jj = p / cnt;
        const int kk = p - jj * cnt;
        if (jj == kk) continue;
        const float dj  = nb_d[jj], dk = nb_d[kk];
        const float dot = nb_x[jj] * nb_x[kk] + nb_y[jj] * nb_y[kk] + nb_z[jj] * nb_z[kk];
        const float ct  = 0.95f * dot / (dj * dk);                 // cos(theta)
        const float st  = sqrtf(fmaxf(1.0f - ct * ct, 0.0f));      // sin(theta) >= 0
        const float w   = nb_w[jj] * nb_w[kk];
        const float davg = 0.5f * (dj + dk);
#pragma unroll
        for (int a4 = 0; a4 < 4; ++a4) {
            const float sa = 0.9f + 0.65f * (float)a4;
            const float da = davg - sa;
            const float ra = w * __expf(-8.0f * da * da);
#pragma unroll
            for (int z = 0; z < 8; ++z) {
                float v   = 0.5f * (1.0f + ct * czv[z] + st * szv[z]);  // ((1+cos(t-s))/2)
                float v2  = v * v;
                float v4  = v2 * v2;
                float v8  = v4 * v4;
                float v16 = v8 * v8;
                float v32 = v16 * v16;                                   // ^zeta, zeta=32
                atomicAdd(&s_ang[a4 * 8 + z], v32 * ra);
            }
        }
    }
    __syncthreads();

    if (t < 16)       out[i * 48 + t] = s_rad[t];
    else if (t < 48)  out[i * 48 + t] = s_ang[t - 16];
}

// ---------------------------------------------------------------------------
extern "C" void kernel_launch(void* const* d_in, const int* in_sizes, int n_in,
                              void* d_out, int out_size, void* d_ws, size_t ws_size,
                              hipStream_t stream) {
    const float* xyz = (const float*)d_in[0];   // [256,3] f32
    const float* q   = (const float*)d_in[1];   // [256]   f32
    float* out  = (float*)d_out;                // [256,48] f32
    float* dist = (float*)d_ws;                 // 256*256 f32 = 256 KB scratch

    ani_dist_wmma<<<dim3(16, 16), 32, 0, stream>>>(xyz, dist);
    ani_aev<<<N_ATOMS, 256, 0, stream>>>(xyz, q, dist, out);
}